// QuantumLayer_29987461660719
// MI455X (gfx1250) — compile-verified
//
#include <hip/hip_runtime.h>
#include <hip/hip_bf16.h>

typedef float v2f __attribute__((ext_vector_type(2)));
typedef float v8f __attribute__((ext_vector_type(8)));

#define WPB 8  // waves per block (256 threads)

// ---------------------------------------------------------------------------
// Kernel 1: build the fixed 16x16 complex unitary of the variational layers,
// then fold the data-encoding phase (-i)^popcount(col) into each column:
//   U'[m][j] = U[m][j] * (-i)^pc(j)
// Output: ws[0..255] = U'_re[m][j] (row-major), ws[256..511] = U'_im[m][j].
// ---------------------------------------------------------------------------
__global__ void build_unitary_kernel(const float* __restrict__ w,
                                     float* __restrict__ U) {
  __shared__ float cre[16][16];
  __shared__ float cim[16][16];
  const int j = threadIdx.x;
  if (j < 16) {
    for (int m = 0; m < 16; ++m) { cre[j][m] = (m == j) ? 1.f : 0.f; cim[j][m] = 0.f; }
    for (int l = 0; l < 2; ++l) {
      for (int q = 0; q < 4; ++q) {
        const float tx = w[(l * 4 + q) * 3 + 0];
        const float ty = w[(l * 4 + q) * 3 + 1];
        const float tz = w[(l * 4 + q) * 3 + 2];
        const int mask = 1 << (3 - q);
        // RX: [[c, -is], [-is, c]]
        {
          float c, s; __sincosf(0.5f * tx, &s, &c);
          for (int i = 0; i < 16; ++i) if (!(i & mask)) {
            const int k2 = i | mask;
            float a0r = cre[j][i],  a0i = cim[j][i];
            float a1r = cre[j][k2], a1i = cim[j][k2];
            cre[j][i]  = c * a0r + s * a1i;
            cim[j][i]  = c * a0i - s * a1r;
            cre[j][k2] = s * a0i + c * a1r;
            cim[j][k2] = -s * a0r + c * a1i;
          }
        }
        // RY: [[c, -s], [s, c]] (real)
        {
          float c, s; __sincosf(0.5f * ty, &s, &c);
          for (int i = 0; i < 16; ++i) if (!(i & mask)) {
            const int k2 = i | mask;
            float a0r = cre[j][i],  a0i = cim[j][i];
            float a1r = cre[j][k2], a1i = cim[j][k2];
            cre[j][i]  = c * a0r - s * a1r;
            cim[j][i]  = c * a0i - s * a1i;
            cre[j][k2] = s * a0r + c * a1r;
            cim[j][k2] = s * a0i + c * a1i;
          }
        }
        // RZ: diag(c - is, c + is)
        {
          float c, s; __sincosf(0.5f * tz, &s, &c);
          for (int i = 0; i < 16; ++i) if (!(i & mask)) {
            const int k2 = i | mask;
            float a0r = cre[j][i],  a0i = cim[j][i];
            float a1r = cre[j][k2], a1i = cim[j][k2];
            cre[j][i]  = c * a0r + s * a0i;
            cim[j][i]  = c * a0i - s * a0r;
            cre[j][k2] = c * a1r - s * a1i;
            cim[j][k2] = c * a1i + s * a1r;
          }
        }
      }
      // CNOT ring: (0,1),(1,2),(2,3),(3,0)
      const int cq[4] = {0, 1, 2, 3}, tq[4] = {1, 2, 3, 0};
      for (int e = 0; e < 4; ++e) {
        const int mc = 1 << (3 - cq[e]), mt = 1 << (3 - tq[e]);
        for (int i = 0; i < 16; ++i) if ((i & mc) && !(i & mt)) {
          const int k2 = i | mt;
          float tr = cre[j][i]; cre[j][i] = cre[j][k2]; cre[j][k2] = tr;
          float ti = cim[j][i]; cim[j][i] = cim[j][k2]; cim[j][k2] = ti;
        }
      }
    }
    // Column j of U is the evolved e_j; fold in (-i)^popcount(j).
    const int pc = __builtin_popcount(j) & 3;
    for (int m = 0; m < 16; ++m) {
      const float r = cre[j][m], i = cim[j][m];
      float rr, ii;
      if      (pc == 0) { rr = r;  ii = i;  }
      else if (pc == 1) { rr = i;  ii = -r; }   // *( -i )
      else if (pc == 2) { rr = -r; ii = -i; }   // *( -1 )
      else              { rr = -i; ii = r;  }   // *(  i )
      U[m * 16 + j]       = rr;
      U[256 + m * 16 + j] = ii;
    }
  }
}

// ---------------------------------------------------------------------------
// Kernel 2: Phi[32,B] = [U'_re ; U'_im] (32x16 real) x Mag[16,B] via
// V_WMMA_F32_16X16X4_F32 (K=16 -> 4 K-steps), then Z-expectations.
// Each wave: 32 samples/iter = two 16-col tiles, 16 WMMAs.
// ---------------------------------------------------------------------------
__global__ __launch_bounds__(256) void qlayer_kernel(
    const float* __restrict__ in, const float* __restrict__ U,
    float* __restrict__ out, int Bn) {
  // Per-wave B tiles: [wave][tile][K=16][N=16] (+1 pad against bank conflicts)
  __shared__ float Bs[WPB][2][16][17];

  const int lane = threadIdx.x & 31;
  const int wib  = threadIdx.x >> 5;
  const int n = lane & 15;   // column / M-row index within tile
  const int h = lane >> 4;   // half select (K pairs / C upper rows)

  // --- Preload A fragments (U' is launch-constant). lane holds row m = n. ---
  // Row-block 0 (-> phi_re): A0 = U'_re ; row-block 1 (-> phi_im): A1 = U'_im
  v2f aR[4], aI[4];
#pragma unroll
  for (int k = 0; k < 4; ++k) {
#pragma unroll
    for (int j = 0; j < 2; ++j) {
      const int kk = 4 * k + 2 * h + j;
      aR[k][j] = U[n * 16 + kk];
      aI[k][j] = U[256 + n * 16 + kk];
    }
  }

  const int wavesTotal = (int)((gridDim.x * blockDim.x) >> 5);
  const int wave       = (int)((blockIdx.x * blockDim.x + threadIdx.x) >> 5);
  const int nGroups    = Bn >> 5;  // B is a multiple of 32

  for (int g = wave; g < nGroups; g += wavesTotal) {
    // ---- Magnitude vector for sample s = g*32 + lane -> LDS column ----
    const int s = (g << 5) + lane;
    const float4 x = ((const float4*)in)[s];
    float c0, s0, c1, s1, c2, s2, c3, s3;
    __sincosf(0.5f * x.x, &s0, &c0);
    __sincosf(0.5f * x.y, &s1, &c1);
    __sincosf(0.5f * x.z, &s2, &c2);
    __sincosf(0.5f * x.w, &s3, &c3);
    float g01[4], g23[4];
    g01[0] = c0 * c1; g01[1] = c0 * s1; g01[2] = s0 * c1; g01[3] = s0 * s1;
    g23[0] = c2 * c3; g23[1] = c2 * s3; g23[2] = s2 * c3; g23[3] = s2 * s3;
#pragma unroll
    for (int idx = 0; idx < 16; ++idx) {
      const int b0 = (idx >> 3) & 1, b1 = (idx >> 2) & 1;
      const int b2 = (idx >> 1) & 1, b3 = idx & 1;
      Bs[wib][h][idx][n] = g01[b0 * 2 + b1] * g23[b2 * 2 + b3];
    }
    // Wave is lockstep; DS is in-order per wave: wait for our LDS stores.
    __builtin_amdgcn_wave_barrier();
    asm volatile("s_wait_dscnt 0" ::: "memory");

    // ---- Two 16-sample tiles: 4 K-steps x {re,im} row blocks ----
#pragma unroll
    for (int t = 0; t < 2; ++t) {
      v2f bf[4];
#pragma unroll
      for (int k = 0; k < 4; ++k) {
        bf[k][0] = Bs[wib][t][4 * k + 2 * h + 0][n];
        bf[k][1] = Bs[wib][t][4 * k + 2 * h + 1][n];
      }
      v8f accR = {};
      v8f accI = {};
#pragma unroll
      for (int k = 0; k < 4; ++k) {
        accR = __builtin_amdgcn_wmma_f32_16x16x4_f32(
            false, aR[k], false, bf[k], (short)0, accR, false, false);
        accI = __builtin_amdgcn_wmma_f32_16x16x4_f32(
            false, aI[k], false, bf[k], (short)0, accI, false, false);
      }
      // ---- Epilogue: lane holds amps m = 8*h + v of sample (t,n) ----
      float p[8];
#pragma unroll
      for (int v = 0; v < 8; ++v)
        p[v] = accR[v] * accR[v] + accI[v] * accI[v];
      // Partial Walsh butterfly: z0=total(sign by h later), z1=bit2, z2=bit1, z3=bit0
      float sA[4], dA[4];
#pragma unroll
      for (int i = 0; i < 4; ++i) { sA[i] = p[2*i] + p[2*i+1]; dA[i] = p[2*i] - p[2*i+1]; }
      const float t01 = sA[0] + sA[1], t23 = sA[2] + sA[3];
      float z0 = t01 + t23;                       // total probability half-sum
      float z1 = t01 - t23;                       // sign on bit2 of m
      float z2 = (sA[0] - sA[1]) + (sA[2] - sA[3]); // sign on bit1
      float z3 = (dA[0] + dA[1]) + (dA[2] + dA[3]); // sign on bit0
      z0 = h ? -z0 : z0;                          // qubit0 sign = bit3 = h
      z0 += __shfl_xor(z0, 16, 32);
      z1 += __shfl_xor(z1, 16, 32);
      z2 += __shfl_xor(z2, 16, 32);
      z3 += __shfl_xor(z3, 16, 32);
      if (h == 0) {
        const int sample = (g << 5) + (t << 4) + n;
        ((float4*)out)[sample] = make_float4(z0, z1, z2, z3);
      }
    }
  }
}

// ---------------------------------------------------------------------------
extern "C" void kernel_launch(void* const* d_in, const int* in_sizes, int n_in,
                              void* d_out, int out_size, void* d_ws, size_t ws_size,
                              hipStream_t stream) {
  const float* inputs  = (const float*)d_in[0];   // [B,4] f32
  const float* weights = (const float*)d_in[1];   // [2,4,3] f32
  float* U   = (float*)d_ws;                      // 512 floats: U'_re | U'_im
  float* out = (float*)d_out;                     // [B,4] f32
  const int Bn = in_sizes[0] / 4;

  build_unitary_kernel<<<dim3(1), dim3(32), 0, stream>>>(weights, U);

  const int nGroups = Bn >> 5;             // 32 samples per wave-iteration
  int blocks = 1024;                        // grid-stride
  const int maxBlocks = (nGroups + WPB - 1) / WPB;
  if (blocks > maxBlocks) blocks = (maxBlocks > 0) ? maxBlocks : 1;
  qlayer_kernel<<<dim3(blocks), dim3(256), 0, stream>>>(inputs, U, out, Bn);
}